// ALNNLayer_21217138442353
// MI455X (gfx1250) — compile-verified
//
#include <hip/hip_runtime.h>

typedef float v2f __attribute__((ext_vector_type(2)));
typedef float v8f __attribute__((ext_vector_type(8)));

#define B_SZ 64
#define K_SZ 13
#define L_SZ 200
#define D_SZ 64

// One block per (b,k). 128 threads = 4 waves; wave w owns d-tile [16w, 16w+16).
// Each wave: loop l in chunks of 4, compute y = w_v * lat in the exact
// 16x4 fp32 WMMA A-layout, reduce over L with V_WMMA_F32_16X16X4_F32
// against an all-ones B matrix, accumulating in the C operand.
__global__ __launch_bounds__(128) void alnn_fused_kernel(
    const float* __restrict__ X,   const float* __restrict__ T,
    const float* __restrict__ M,   const float* __restrict__ DT,
    const float* __restrict__ alpha, const float* __restrict__ w_v,
    const float* __restrict__ w_t, const float* __restrict__ b_v,
    const float* __restrict__ b_t, float* __restrict__ out)
{
    const int bk   = blockIdx.x;          // 0 .. B*K-1
    const int b    = bk / K_SZ;
    const int k    = bk - b * K_SZ;
    const int wave = threadIdx.x >> 5;    // 0..3  -> d tile
    const int lane = threadIdx.x & 31;
    const int m    = lane & 15;           // row (d within tile)
    const int hi   = lane >> 4;           // 0: K=0,1   1: K=2,3
    const int d    = wave * 16 + m;

    const float a_k = fmaxf(alpha[k], 0.0f);
    const float ref = 4.0f * (float)k;    // linspace(0,48,13)[k] exactly

    const float* __restrict__ Xb  = X  + (size_t)b * L_SZ * D_SZ;
    const float* __restrict__ Tb  = T  + (size_t)b * L_SZ * D_SZ;
    const float* __restrict__ Mb  = M  + (size_t)b * L_SZ * D_SZ;
    const float* __restrict__ DTb = DT + (size_t)b * L_SZ * D_SZ;
    const int wbase = k * L_SZ * D_SZ;    // base into [K,L,D] weights

    v8f acc = {0.f, 0.f, 0.f, 0.f, 0.f, 0.f, 0.f, 0.f};
    v2f ones; ones.x = 1.0f; ones.y = 1.0f;

    for (int l0 = 0; l0 < L_SZ; l0 += 4) {
        v2f A;
        #pragma unroll
        for (int c = 0; c < 2; ++c) {
            const int l    = l0 + 2 * hi + c;      // this lane's K-slot -> l
            const int idx  = l * D_SZ + d;         // [L,D] slice index
            const int widx = wbase + idx;          // [K,L,D] weight index

            const float x  = Xb[idx];
            const float t  = Tb[idx];
            const float mm = Mb[idx];
            const float dt = DTb[idx];

            const float kern  = __expf(-a_k * fabsf(t - ref));
            const float inten = fmaxf(x * kern, 0.0f);

            const float4 wt = *(const float4*)(w_t + (size_t)widx * 4);
            float lat = fmaf(wt.x, x,
                        fmaf(wt.y, dt,
                        fmaf(wt.z, inten,
                        fmaf(wt.w, mm, 4.0f * b_t[widx]))));
            lat = fmaxf(lat, 0.0f);

            const float y = w_v[widx] * lat;
            if (c == 0) A.x = y; else A.y = y;
        }
        // D[m,n] = sum_{kk<4} A[m,kk] * 1 + C  -> row sums, accumulated over L
        acc = __builtin_amdgcn_wmma_f32_16x16x4_f32(
            /*neg_a=*/false, A, /*neg_b=*/false, ones,
            /*c_mod=*/(short)0, acc, /*reuse_a=*/false, /*reuse_b=*/false);
    }

    // C/D layout: vgpr r, lanes 0-15 -> (M=r,   N=lane),
    //                     lanes 16-31-> (M=r+8, N=lane-16).
    // Column N=0 lives in lane 0 (M=0..7) and lane 16 (M=8..15).
    if (m == 0) {
        const int dbase = wave * 16 + hi * 8;
        float* __restrict__ o = out + ((size_t)b * K_SZ + k) * D_SZ + dbase;
        const float* __restrict__ bv = b_v + k * D_SZ + dbase;
        #pragma unroll
        for (int r = 0; r < 8; ++r) {
            const float v = acc[r] + 200.0f * bv[r];  // b_v broadcast over L
            o[r] = fmaxf(v, 0.0f);
        }
    }
}

extern "C" void kernel_launch(void* const* d_in, const int* in_sizes, int n_in,
                              void* d_out, int out_size, void* d_ws, size_t ws_size,
                              hipStream_t stream) {
    const float* X     = (const float*)d_in[0];
    const float* T     = (const float*)d_in[1];
    const float* M     = (const float*)d_in[2];
    const float* DT    = (const float*)d_in[3];
    const float* alpha = (const float*)d_in[4];
    const float* w_v   = (const float*)d_in[5];
    const float* w_t   = (const float*)d_in[6];
    const float* b_v   = (const float*)d_in[7];
    const float* b_t   = (const float*)d_in[8];
    float* out = (float*)d_out;

    dim3 grid(B_SZ * K_SZ);   // 832 blocks, one per (b,k)
    dim3 block(128);          // 4 waves, one per 16-wide d-tile
    alnn_fused_kernel<<<grid, block, 0, stream>>>(X, T, M, DT, alpha,
                                                  w_v, w_t, b_v, b_t, out);
}